// SparseLinear_48189533061453
// MI455X (gfx1250) — compile-verified
//
#include <hip/hip_runtime.h>
#include <stdint.h>

#define WAVES_PER_BLOCK 8
#define CHUNK 32

// ---------------- CDNA5 async-to-LDS plumbing ----------------
#if defined(__gfx1250__) && __has_builtin(__builtin_amdgcn_s_wait_asynccnt)
#define WAIT_ASYNC(n) do { __builtin_amdgcn_s_wait_asynccnt(n); asm volatile("" ::: "memory"); } while (0)
#else
#define WAIT_ASYNC(n) asm volatile("s_wait_asynccnt %0" :: "n"(n) : "memory")
#endif

#if defined(__gfx1250__) && __has_builtin(__builtin_amdgcn_global_load_async_to_lds_b32)
typedef __attribute__((address_space(1))) int* gas1_iptr;
typedef __attribute__((address_space(3))) int* gas3_iptr;
// builtin signature (probe-confirmed): (global int*, lds int*, imm offset, imm cpol)
#define ASYNC_LOAD_B32(g, l) \
  __builtin_amdgcn_global_load_async_to_lds_b32((gas1_iptr)(g), (gas3_iptr)(l), 0, 0)
#else
__device__ __forceinline__ void async_load_b32_asm(const void* g, void* l) {
  unsigned lds_addr = (unsigned)(uintptr_t)l;  // low 32 bits of generic addr = LDS offset
  asm volatile("global_load_async_to_lds_b32 %0, %1, off"
               :: "v"(lds_addr), "v"(g) : "memory");
}
#define ASYNC_LOAD_B32(g, l) async_load_b32_asm((g), (l))
#endif

// ---------------- HOT kernel first (so the disasm snippet shows the async path) ----------------
__device__ __forceinline__ void process_edge(int4 t, int lane,
                                             const float* __restrict__ xT,
                                             float* __restrict__ outT) {
  // t is wave-uniform (all lanes loaded the same LDS address) -> scalarize
  const int s = __builtin_amdgcn_readfirstlane(t.x);
  const int d = __builtin_amdgcn_readfirstlane(t.y);
  const float v = __int_as_float(__builtin_amdgcn_readfirstlane(t.z));
  const float contrib = v * xT[(size_t)s * 32 + lane];   // one 128B cacheline per edge
  atomicAdd(&outT[(size_t)d * 32 + lane], contrib);      // one 128B f32 atomic per edge
}

__device__ __forceinline__ void prefetch_chunk(const int* __restrict__ src,
                                               const int* __restrict__ dst,
                                               const float* __restrict__ val,
                                               int e, int nnz, int4* slot) {
  if (e < nnz) {
    int* l = (int*)slot;
    ASYNC_LOAD_B32(src + e, l + 0);
    ASYNC_LOAD_B32(dst + e, l + 1);
    ASYNC_LOAD_B32(val + e, l + 2);
  }
}

__global__ __launch_bounds__(256) void k_edge_scatter(const int* __restrict__ src,
                                                      const int* __restrict__ dst,
                                                      const float* __restrict__ val,
                                                      const float* __restrict__ xT,
                                                      float* __restrict__ outT, int nnz) {
  // per-wave double buffer of 32 edge slots, padded to int4 for ds_load_b128 readback
  __shared__ int4 stage[WAVES_PER_BLOCK][2][CHUNK];
  const int lane = threadIdx.x & 31;
  const int wib = threadIdx.x >> 5;
  const int gwave = blockIdx.x * WAVES_PER_BLOCK + wib;
  const int nWaves = gridDim.x * WAVES_PER_BLOCK;
  const int nChunks = (nnz + CHUNK - 1) / CHUNK;

  int bufSel = 0;
  if (gwave < nChunks) {
    prefetch_chunk(src, dst, val, gwave * CHUNK + lane, nnz, &stage[wib][0][lane]);
  }

  for (int c = gwave; c < nChunks; c += nWaves) {
    const int cn = c + nWaves;
    if (cn < nChunks) {
      prefetch_chunk(src, dst, val, cn * CHUNK + lane, nnz, &stage[wib][bufSel ^ 1][lane]);
      WAIT_ASYNC(3);  // async loads complete in order: current chunk's 3 are done
    } else {
      WAIT_ASYNC(0);
    }

    const int4* b = stage[wib][bufSel];
    const int kmax = (nnz - c * CHUNK >= CHUNK) ? CHUNK : (nnz - c * CHUNK);
    if (kmax == CHUNK) {
#pragma unroll 8
      for (int k = 0; k < CHUNK; ++k) {
        process_edge(b[k], lane, xT, outT);  // uniform-address ds_load_b128 broadcast
      }
    } else {
      for (int k = 0; k < kmax; ++k) {
        process_edge(b[k], lane, xT, outT);
      }
    }
    bufSel ^= 1;
  }
}

// ---------------- x (32, N_in) -> xT (N_in, 32) ----------------
__global__ __launch_bounds__(256) void k_transpose_x(const float* __restrict__ x,
                                                     float* __restrict__ xT, int n_in) {
  __shared__ float tile[32][33];
  const int i0 = blockIdx.x * 32;
  const int lx = threadIdx.x & 31;
  const int ly = threadIdx.x >> 5;  // 0..7
#pragma unroll
  for (int r = 0; r < 32; r += 8) {
    const int b = ly + r;
    const int i = i0 + lx;
    tile[b][lx] = (i < n_in) ? x[(size_t)b * n_in + i] : 0.0f;
  }
  __syncthreads();
#pragma unroll
  for (int r = 0; r < 32; r += 8) {
    const int i = i0 + ly + r;
    if (i < n_in) xT[(size_t)i * 32 + lx] = tile[lx][ly + r];
  }
}

// ---------------- outT[j*32+b] = bias[j] ----------------
__global__ __launch_bounds__(256) void k_init_outT(const float* __restrict__ bias,
                                                   float* __restrict__ outT, int n_out) {
  const int idx = blockIdx.x * 256 + threadIdx.x;
  if (idx < n_out * 32) outT[idx] = bias[idx >> 5];
}

// ---------------- outT (N_out, 32) -> out (32, N_out) ----------------
__global__ __launch_bounds__(256) void k_write_out(const float* __restrict__ outT,
                                                   float* __restrict__ out, int n_out) {
  __shared__ float tile[32][33];
  const int j0 = blockIdx.x * 32;
  const int lx = threadIdx.x & 31;
  const int ly = threadIdx.x >> 5;
#pragma unroll
  for (int r = 0; r < 32; r += 8) {
    const int j = j0 + ly + r;
    if (j < n_out) tile[ly + r][lx] = outT[(size_t)j * 32 + lx];
  }
  __syncthreads();
#pragma unroll
  for (int r = 0; r < 32; r += 8) {
    const int b = ly + r;
    const int j = j0 + lx;
    if (j < n_out) out[(size_t)b * n_out + j] = tile[lx][b];
  }
}

// ---------------- Fallback (workspace too small): direct strided path ----------------
__global__ __launch_bounds__(256) void k_init_out_direct(const float* __restrict__ bias,
                                                         float* __restrict__ out, int n_out) {
  const int idx = blockIdx.x * 256 + threadIdx.x;
  if (idx < 32 * n_out) out[idx] = bias[idx % n_out];
}

__global__ __launch_bounds__(256) void k_edge_direct(const int* __restrict__ src,
                                                     const int* __restrict__ dst,
                                                     const float* __restrict__ val,
                                                     const float* __restrict__ x,
                                                     float* __restrict__ out,
                                                     int nnz, int n_in, int n_out) {
  const int lane = threadIdx.x & 31;
  const int gwave = (blockIdx.x * 256 + (int)threadIdx.x) >> 5;
  const int nWaves = (gridDim.x * 256) >> 5;
  const int nChunks = (nnz + 31) >> 5;
  for (int c = gwave; c < nChunks; c += nWaves) {
    const int e = c * 32 + lane;
    int s = 0, d = 0;
    float v = 0.0f;
    if (e < nnz) { s = src[e]; d = dst[e]; v = val[e]; }
    for (int k = 0; k < 32; ++k) {
      const int ss = __shfl(s, k, 32);
      const int dd = __shfl(d, k, 32);
      const float vv = __shfl(v, k, 32);
      const float contrib = vv * x[(size_t)lane * n_in + ss];
      atomicAdd(&out[(size_t)lane * n_out + dd], contrib);
    }
  }
}

// ---------------- Launcher ----------------
extern "C" void kernel_launch(void* const* d_in, const int* in_sizes, int n_in_args,
                              void* d_out, int out_size, void* d_ws, size_t ws_size,
                              hipStream_t stream) {
  (void)n_in_args;
  const float* x      = (const float*)d_in[0];
  const float* values = (const float*)d_in[1];
  const float* bias   = (const float*)d_in[2];
  const int*   indices = (const int*)d_in[3];
  // d_in[4] is n_out as a device scalar; sizes come from in_sizes instead.
  const int nnz   = in_sizes[1];
  const int n_out = in_sizes[2];
  const int B     = 32;
  const int n_in  = in_sizes[0] / B;
  const int* srcIdx = indices;          // indices[0, :]
  const int* dstIdx = indices + nnz;    // indices[1, :]
  float* out = (float*)d_out;
  (void)out_size;

  const size_t need = (size_t)(n_in + n_out) * 32 * sizeof(float);
  if (ws_size >= need) {
    float* xT   = (float*)d_ws;
    float* outT = xT + (size_t)n_in * 32;
    k_transpose_x<<<(n_in + 31) / 32, 256, 0, stream>>>(x, xT, n_in);
    k_init_outT<<<(n_out * 32 + 255) / 256, 256, 0, stream>>>(bias, outT, n_out);
    k_edge_scatter<<<2048, 256, 0, stream>>>(srcIdx, dstIdx, values, xT, outT, nnz);
    k_write_out<<<(n_out + 31) / 32, 256, 0, stream>>>(outT, out, n_out);
  } else {
    k_init_out_direct<<<(32 * n_out + 255) / 256, 256, 0, stream>>>(bias, out, n_out);
    k_edge_direct<<<1024, 256, 0, stream>>>(srcIdx, dstIdx, values, x, out, nnz, n_in, n_out);
  }
}